// MSBlending_89077621719753
// MI455X (gfx1250) — compile-verified
//
#include <hip/hip_runtime.h>
#include <hip/hip_bf16.h>

typedef float v2f __attribute__((ext_vector_type(2)));
typedef float v8f __attribute__((ext_vector_type(8)));

#define HW   (768 * 768)          // 589824
#define NPTS 294912
#define NF   48
#define K1   49                   // F + 1 (concat of s row)
#define STR  132                  // LDS row stride (floats), padded vs 128

#define WMMA_F32_4(acc, a, b) \
    acc = __builtin_amdgcn_wmma_f32_16x16x4_f32(false, (a), false, (b), (short)0, (acc), false, false)

// ---------------------------------------------------------------------------
// Kernel 0: zero the (3, HW) scatter target in scratch
// ---------------------------------------------------------------------------
__global__ __launch_bounds__(256) void zero_kernel(float* __restrict__ w, int n) {
    int i = blockIdx.x * 256 + threadIdx.x;
    if (i < n) w[i] = 0.0f;
}

// ---------------------------------------------------------------------------
// Kernel 1: gather + per-point MLP (WMMA f32 16x16x4) + scatter weights
// Block = 256 threads = 8 waves; each wave owns 16 points (its 16 LDS columns).
// ---------------------------------------------------------------------------
__global__ __launch_bounds__(256) void mlp_scatter_kernel(
    const float* __restrict__ c,     // (48, HW)
    const float* __restrict__ jm,    // (N, 2, 2)
    const int*   __restrict__ yi,    // (N)
    const float* __restrict__ W1, const float* __restrict__ b1,   // (48,49),(48)
    const float* __restrict__ W2, const float* __restrict__ b2,   // (48,48),(48)
    const float* __restrict__ W3, const float* __restrict__ b3,   // (3,48),(3)
    float* __restrict__ wout)        // (3, HW), pre-zeroed
{
    __shared__ float xbuf[52 * STR];   // layer input, [K][n] layout (rows 49..51 = pad)
    __shared__ float hbuf[48 * STR];   // hidden activations
    __shared__ int   idx[128];

    const int tid  = threadIdx.x;
    const int base = blockIdx.x * 128;           // N == 2304 * 128 exactly

    // ---- stage point indices -------------------------------------------------
    if (tid < 128) idx[tid] = yi[base + tid];
    __syncthreads();

    // ---- cooperative gather: c -> xbuf rows 0..47 ---------------------------
    #pragma unroll
    for (int it = 0; it < 24; ++it) {
        int e = tid + 256 * it;                  // 0..6143
        int k = e >> 7;                          // 0..47
        int n = e & 127;
        xbuf[k * STR + n] = c[k * HW + idx[n]];
    }
    // ---- s row (K=48) and zero pad rows 49..51 ------------------------------
    if (tid < 128) {
        const float* jp = jm + 4 * (base + tid);
        float det = jp[0] * jp[3] - jp[1] * jp[2];
        xbuf[48 * STR + tid] = -logf(fabsf(det) + 1e-8f);
        xbuf[49 * STR + tid] = 0.0f;
        xbuf[50 * STR + tid] = 0.0f;
        xbuf[51 * STR + tid] = 0.0f;
    }
    __syncthreads();

    // ---- per-wave WMMA fragment coordinates ---------------------------------
    const int lane = tid & 31;
    const int wave = tid >> 5;
    const int n    = lane & 15;        // N index within tile / A-row index
    const int kh   = lane >> 4;        // K half: 0 -> K0/K1, 1 -> K2/K3
    const int col  = wave * 16 + n;    // this wave's private LDS column

    // ======================= layer 1: 48x49 @ 49x16 =========================
    #pragma unroll
    for (int mt = 0; mt < 3; ++mt) {
        v8f acc = {};
        const int row = mt * 16 + n;   // output row 0..47
        #pragma unroll
        for (int k0 = 0; k0 < 52; k0 += 4) {
            int kA = k0 + 2 * kh;
            v2f a, b;
            a.x = (kA     < K1) ? W1[row * K1 + kA]     : 0.0f;
            a.y = (kA + 1 < K1) ? W1[row * K1 + kA + 1] : 0.0f;
            b.x = xbuf[kA * STR + col];            // pad rows are zeroed
            b.y = xbuf[(kA + 1) * STR + col];
            WMMA_F32_4(acc, a, b);
        }
        #pragma unroll
        for (int v = 0; v < 8; ++v) {              // D: VGPR v -> M = mt*16 + v + 8*kh
            int M = mt * 16 + v + 8 * kh;
            hbuf[M * STR + col] = fmaxf(acc[v] + b1[M], 0.0f);
        }
    }

    // ======================= layer 2: 48x48 @ 48x16 =========================
    #pragma unroll
    for (int mt = 0; mt < 3; ++mt) {
        v8f acc = {};
        const int row = mt * 16 + n;
        #pragma unroll
        for (int k0 = 0; k0 < 48; k0 += 4) {
            int kA = k0 + 2 * kh;
            v2f a, b;
            a.x = W2[row * NF + kA];
            a.y = W2[row * NF + kA + 1];
            b.x = hbuf[kA * STR + col];
            b.y = hbuf[(kA + 1) * STR + col];
            WMMA_F32_4(acc, a, b);
        }
        #pragma unroll
        for (int v = 0; v < 8; ++v) {
            int M = mt * 16 + v + 8 * kh;
            xbuf[M * STR + col] = fmaxf(acc[v] + b2[M], 0.0f);  // reuse xbuf rows 0..47
        }
    }

    // ======================= layer 3: 3x48 @ 48x16 ==========================
    v8f acc3 = {};
    #pragma unroll
    for (int k0 = 0; k0 < 48; k0 += 4) {
        int kA = k0 + 2 * kh;
        v2f a, b;
        a.x = (n < 3) ? W3[n * NF + kA]     : 0.0f;   // rows 3..15 of A are zero
        a.y = (n < 3) ? W3[n * NF + kA + 1] : 0.0f;
        b.x = xbuf[kA * STR + col];
        b.y = xbuf[(kA + 1) * STR + col];
        WMMA_F32_4(acc3, a, b);
    }

    // ---- scatter: lanes 0..15 hold M = 0..7 in acc3[0..7]; we need M=0..2 ---
    if (kh == 0) {
        int gi = idx[col];
        wout[0 * HW + gi] = acc3[0] + b3[0];
        wout[1 * HW + gi] = acc3[1] + b3[1];
        wout[2 * HW + gi] = acc3[2] + b3[2];
    }
}

// ---------------------------------------------------------------------------
// Kernel 2: blend  out[c,hw] = sum_p w[p,hw] * ws[p,c,hw]   (HBM-bound stream)
// grid: (HW/256, 4); each thread: 1 pixel x 16 channels
// ---------------------------------------------------------------------------
__global__ __launch_bounds__(256) void blend_kernel(
    const float* __restrict__ ws,    // (3, 64, HW)
    const float* __restrict__ w,     // (3, HW)
    float* __restrict__ out)         // (64, HW)
{
    const int i  = blockIdx.x * 256 + threadIdx.x;   // pixel
    const int c0 = blockIdx.y * 16;                  // channel chunk

    const float w0 = w[i];
    const float w1 = w[HW + i];
    const float w2 = w[2 * HW + i];

    #pragma unroll 4
    for (int cc = 0; cc < 16; ++cc) {
        size_t o = (size_t)(c0 + cc) * HW + i;
        out[o] = w0 * ws[o]
               + w1 * ws[(size_t)64  * HW + o]
               + w2 * ws[(size_t)128 * HW + o];
    }
}

// ---------------------------------------------------------------------------
extern "C" void kernel_launch(void* const* d_in, const int* in_sizes, int n_in,
                              void* d_out, int out_size, void* d_ws, size_t ws_size,
                              hipStream_t stream) {
    const float* ws = (const float*)d_in[0];
    const float* c  = (const float*)d_in[1];
    const float* jm = (const float*)d_in[2];
    const int*   yi = (const int*)  d_in[3];
    const float* W1 = (const float*)d_in[4];
    const float* b1 = (const float*)d_in[5];
    const float* W2 = (const float*)d_in[6];
    const float* b2 = (const float*)d_in[7];
    const float* W3 = (const float*)d_in[8];
    const float* b3 = (const float*)d_in[9];
    float* out  = (float*)d_out;
    float* wbuf = (float*)d_ws;                      // (3, HW) floats = ~6.75 MB

    const int nW = 3 * HW;                           // 1,769,472
    zero_kernel<<<(nW + 255) / 256, 256, 0, stream>>>(wbuf, nW);

    mlp_scatter_kernel<<<NPTS / 128, 256, 0, stream>>>(
        c, jm, yi, W1, b1, W2, b2, W3, b3, wbuf);

    dim3 bgrid(HW / 256, 4);
    blend_kernel<<<bgrid, 256, 0, stream>>>(ws, wbuf, out);
}